// GNNTransformerModel_51848845197366
// MI455X (gfx1250) — compile-verified
//
#include <hip/hip_runtime.h>
#include <math.h>

// ---- problem constants (from reference) ----
#define B_    8
#define T_    64
#define D_    128
#define H_    4
#define L_    4
#define G_    (B_*T_)        // 512
#define NPG_  200
#define N_    (G_*NPG_)      // 102400
#define E_    (8*N_)         // 819200
#define FIN_  2
#define HID_  64
#define DH_   (D_/H_)        // 32
#define NCLS_ 2

typedef __attribute__((ext_vector_type(2))) float v2f;
typedef __attribute__((ext_vector_type(8))) float v8f;

static inline int cdiv(long long a, long long b) { return (int)((a + b - 1) / b); }

// ============================================================
// WMMA f32 GEMM: out = A[M,KT] * B (+bias)(+relu)
//   TRANSB=0: B stored (KT, Nout) row-major  (out = A @ B)
//   TRANSB=1: B stored (Nout, KT) row-major  (out = A @ B^T)
//   SELFLOOP=1 (GCN mode): writes C[m,n]  = val * dinv[m]      (scaled messages)
//                          and    C2[m,n] = val * dinv[m]^2    (self-loop init)
// Block = 4 waves sharing one 16-wide B column panel staged in LDS
// (KT x 16 floats, dynamic smem). Requires tilesM % 4 == 0 and
// KT % 16 == 0 -> no guards, EXEC all-ones for WMMA (ISA 7.12).
// Fragment layouts per CDNA5 ISA 7.12.2 (16x16x4 f32).
// ============================================================
template<int TRANSB, int RELU, int BIAS, int SELFLOOP, int KT>
__global__ __launch_bounds__(128) void k_wmma_gemm(
    const float* __restrict__ A, const float* __restrict__ Bm,
    const float* __restrict__ bias, const float* __restrict__ dinv,
    float* __restrict__ C, float* __restrict__ C2,
    int tilesMd4, int Nout, int lda, int ldc)
{
  extern __shared__ float Bs[];                 // [KT][16]
  int tn  = blockIdx.x / tilesMd4;
  int tm  = (blockIdx.x - tn * tilesMd4) * 4 + (threadIdx.x >> 5);
  int n0  = tn << 4;
  int w   = threadIdx.x >> 5;
  int ln  = threadIdx.x & 31;

  // ---- stage B column panel into LDS (coalesced) ----
  if (TRANSB) {
    // B(n,k) at Bm[n*KT + k]; each wave stages 4 of the 16 columns
    for (int nl = w * 4; nl < w * 4 + 4; ++nl) {
      const float* src = Bm + (size_t)(n0 + nl) * KT;
      for (int k = ln; k < KT; k += 32) Bs[(k << 4) + nl] = src[k];
    }
  } else {
    // B(k,n) at Bm[k*Nout + n]
    for (int idx = threadIdx.x; idx < (KT << 4); idx += 128) {
      int k = idx >> 4, nl = idx & 15;
      Bs[idx] = Bm[(size_t)k * Nout + n0 + nl];
    }
  }
  __syncthreads();

  int lane16 = ln & 15;
  int koff   = (ln >> 4) << 1;                  // 0 (lanes 0-15) / 2 (lanes 16-31)
  const float* Arow = A + (size_t)((tm << 4) + lane16) * lda;

  v8f acc = {};
#pragma unroll 4
  for (int k0 = 0; k0 < KT; k0 += 4) {
    int ka = k0 + koff;
    v2f a, b;
    a[0] = Arow[ka];
    a[1] = Arow[ka + 1];
    b[0] = Bs[(ka << 4) + lane16];
    b[1] = Bs[((ka + 1) << 4) + lane16];
    acc = __builtin_amdgcn_wmma_f32_16x16x4_f32(
        /*neg_a=*/false, a, /*neg_b=*/false, b,
        /*c_mod=*/(short)0, acc, /*reuse_a=*/false, /*reuse_b=*/false);
  }

  int nB    = n0 + lane16;
  int mrow0 = (tm << 4) + ((ln >> 4) << 3);     // rows 0..7 / 8..15 of the tile
  if (SELFLOOP) {
#pragma unroll
    for (int v = 0; v < 8; ++v) {
      int m = mrow0 + v;
      float di  = dinv[m];
      float val = acc[v] * di;
      C [(size_t)m * ldc + nB] = val;           // t * dinv[m]   (message)
      C2[(size_t)m * ldc + nB] = val * di;      // t * dinv[m]^2 (self loop)
    }
  } else {
    float bv = BIAS ? bias[nB] : 0.0f;
#pragma unroll
    for (int v = 0; v < 8; ++v) {
      float val = acc[v] + bv;
      if (RELU) val = fmaxf(val, 0.0f);
      C[(size_t)(mrow0 + v) * ldc + nB] = val;
    }
  }
}

// ============================================================
// GCN support kernels
// ============================================================
__global__ void k_fill(float* __restrict__ p, float v, int n) {
  int i = blockIdx.x * blockDim.x + threadIdx.x;
  if (i < n) p[i] = v;
}

__global__ void k_deg_count(const int* __restrict__ dst, float* __restrict__ deg) {
  int e = blockIdx.x * blockDim.x + threadIdx.x;
  if (e < E_) atomicAdd(&deg[dst[e]], 1.0f);
}

__global__ void k_rsqrt_inplace(float* __restrict__ p, int n) {
  int i = blockIdx.x * blockDim.x + threadIdx.x;
  if (i < n) p[i] = rsqrtf(p[i]);
}

// layer-1 transform (K=FIN=2, scalar) fused with dinv scaling + self-loop init
__global__ void k_gcn1_transform(const float* __restrict__ x,
                                 const float* __restrict__ W,
                                 const float* __restrict__ dinv,
                                 float* __restrict__ t,   // t*dinv (messages)
                                 float* __restrict__ h) { // t*dinv^2 (self loop)
  int i = blockIdx.x * blockDim.x + threadIdx.x;          // N*64
  if (i >= N_ * HID_) return;
  int n = i >> 6, j = i & 63;
  float di = dinv[n];
  float v = (x[n * 2] * W[j] + x[n * 2 + 1] * W[HID_ + j]) * di;
  t[i] = v;
  h[i] = v * di;
}

// out[dst,c] += tscaled[src,c] * dinv[dst]   (one thread per (edge,channel))
__global__ void k_edge_scatter(const float* __restrict__ t,
                               const int* __restrict__ src,
                               const int* __restrict__ dst,
                               const float* __restrict__ dinv,
                               float* __restrict__ out, int shift) {
  long long idx = (long long)blockIdx.x * blockDim.x + threadIdx.x;
  int e = (int)(idx >> shift);
  if (e >= E_) return;
  int c = (int)(idx & ((1 << shift) - 1));
  int s = src[e], d = dst[e];
  atomicAdd(&out[((size_t)d << shift) + c], t[((size_t)s << shift) + c] * dinv[d]);
}

// per-node: h += bias; optionally relu then LayerNorm.  One block per node.
__global__ void k_node_post(float* __restrict__ h, const float* __restrict__ bias,
                            const float* __restrict__ g, const float* __restrict__ bt,
                            int C, int doLN) {
  int r = blockIdx.x, c = threadIdx.x;
  size_t idx = (size_t)r * C + c;
  float v = h[idx] + bias[c];
  if (doLN) {
    v = fmaxf(v, 0.0f);
    __shared__ float s1[128], s2[128];
    s1[c] = v; s2[c] = v * v;
    __syncthreads();
    for (int off = C >> 1; off > 0; off >>= 1) {
      if (c < off) { s1[c] += s1[c + off]; s2[c] += s2[c + off]; }
      __syncthreads();
    }
    float m = s1[0] / C;
    float var = s2[0] / C - m * m;
    v = (v - m) * rsqrtf(var + 1e-5f) * g[c] + bt[c];
  }
  h[idx] = v;
}

// per-graph mean readout fused with sinusoidal PE. One block per graph (128 thr).
__global__ void k_readout_pe(const float* __restrict__ h, float* __restrict__ seq) {
  int gidx = blockIdx.x;        // 0..511 -> (b = g/T, t = g%T)
  int d = threadIdx.x;          // 0..127
  float acc = 0.0f;
  const float* base = h + (size_t)gidx * NPG_ * D_ + d;
  for (int i = 0; i < NPG_; ++i) acc += base[(size_t)i * D_];
  acc *= (1.0f / NPG_);
  int tpos = gidx & (T_ - 1);
  int j = d >> 1;
  float div = __expf((float)(2 * j) * (-logf(10000.0f) / (float)D_));
  float ang = (float)tpos * div;
  float pe = (d & 1) ? __cosf(ang) : __sinf(ang);
  seq[(size_t)gidx * D_ + d] = acc + pe;
}

// ============================================================
// Transformer kernels
// ============================================================
// one block per (b,h); 64 threads = one query row each
__global__ __launch_bounds__(64) void k_attn(const float* __restrict__ qkv,
                                             float* __restrict__ out) {
  int bh = blockIdx.x;
  int b = bh >> 2, h = bh & 3;
  int t = threadIdx.x;                       // query index 0..63
  __shared__ float Ks[T_][DH_];
  __shared__ float Vs[T_][DH_];
  __shared__ float Sm[T_][T_ + 1];
  const float* base = qkv + (size_t)(b * T_ + t) * (3 * D_);
#pragma unroll
  for (int d = 0; d < DH_; ++d) {
    Ks[t][d] = base[D_     + h * DH_ + d];
    Vs[t][d] = base[2 * D_ + h * DH_ + d];
  }
  float q[DH_];
#pragma unroll
  for (int d = 0; d < DH_; ++d) q[d] = base[h * DH_ + d];
  __syncthreads();

  const float scale = 0.17677669529663687f;  // 1/sqrt(32)
  float mx = -1e30f;
  for (int k = 0; k < T_; ++k) {
    float dot = 0.0f;
#pragma unroll
    for (int d = 0; d < DH_; ++d) dot += q[d] * Ks[k][d];
    dot *= scale;
    Sm[t][k] = dot;
    mx = fmaxf(mx, dot);
  }
  float sum = 0.0f;
  for (int k = 0; k < T_; ++k) {
    float e = __expf(Sm[t][k] - mx);
    Sm[t][k] = e;
    sum += e;
  }
  float inv = 1.0f / sum;
  float acc[DH_] = {};
  for (int k = 0; k < T_; ++k) {
    float p = Sm[t][k] * inv;
#pragma unroll
    for (int d = 0; d < DH_; ++d) acc[d] += p * Vs[k][d];
  }
  float* o = out + (size_t)(b * T_ + t) * D_ + h * DH_;
#pragma unroll
  for (int d = 0; d < DH_; ++d) o[d] = acc[d];
}

// seq = LN(seq + delta); one block (128 threads) per row
__global__ void k_resid_ln(float* __restrict__ seq, const float* __restrict__ delta,
                           const float* __restrict__ g, const float* __restrict__ bt) {
  int r = blockIdx.x, c = threadIdx.x;
  size_t idx = (size_t)r * D_ + c;
  float v = seq[idx] + delta[idx];
  __shared__ float s1[D_], s2[D_];
  s1[c] = v; s2[c] = v * v;
  __syncthreads();
  for (int off = D_ >> 1; off > 0; off >>= 1) {
    if (c < off) { s1[c] += s1[c + off]; s2[c] += s2[c + off]; }
    __syncthreads();
  }
  float m = s1[0] * (1.0f / D_);
  float var = s2[0] * (1.0f / D_) - m * m;
  seq[idx] = (v - m) * rsqrtf(var + 1e-5f) * g[c] + bt[c];
}

// temporal mean pool + MLP head; one block per batch element
__global__ void k_head(const float* __restrict__ seq,
                       const float* __restrict__ hW1, const float* __restrict__ hb1,
                       const float* __restrict__ hW2, const float* __restrict__ hb2,
                       float* __restrict__ out) {
  int b = blockIdx.x, tid = threadIdx.x;     // 128 threads
  __shared__ float pooled[D_];
  __shared__ float hh[D_ / 2];
  float acc = 0.0f;
  for (int t = 0; t < T_; ++t) acc += seq[(size_t)(b * T_ + t) * D_ + tid];
  pooled[tid] = acc * (1.0f / T_);
  __syncthreads();
  if (tid < D_ / 2) {
    float a = hb1[tid];
    for (int d = 0; d < D_; ++d) a += pooled[d] * hW1[tid * D_ + d];
    hh[tid] = fmaxf(a, 0.0f);
  }
  __syncthreads();
  if (tid < NCLS_) {
    float a = hb2[tid];
    for (int j = 0; j < D_ / 2; ++j) a += hh[j] * hW2[tid * (D_ / 2) + j];
    out[b * NCLS_ + tid] = a;
  }
}

// ============================================================
// launch
// ============================================================
extern "C" void kernel_launch(void* const* d_in, const int* in_sizes, int n_in,
                              void* d_out, int out_size, void* d_ws, size_t ws_size,
                              hipStream_t stream) {
  const float* x    = (const float*)d_in[0];
  const int*   esrc = (const int*)d_in[1];
  const int*   edst = (const int*)d_in[2];
  // d_in[3] = node_graph (node i -> i/NPG; recomputed analytically)
  const float* gW1 = (const float*)d_in[4];
  const float* gb1 = (const float*)d_in[5];
  const float* ln1g = (const float*)d_in[6];
  const float* ln1b = (const float*)d_in[7];
  const float* gW2 = (const float*)d_in[8];
  const float* gb2 = (const float*)d_in[9];
  const float* ln2g = (const float*)d_in[10];
  const float* ln2b = (const float*)d_in[11];
  const float* gW3 = (const float*)d_in[12];
  const float* gb3 = (const float*)d_in[13];
  const float* tWqkv = (const float*)d_in[14];
  const float* tbqkv = (const float*)d_in[15];
  const float* tWo = (const float*)d_in[16];
  const float* tbo = (const float*)d_in[17];
  const float* tg1 = (const float*)d_in[18];
  const float* tb1 = (const float*)d_in[19];
  const float* tg2 = (const float*)d_in[20];
  const float* tb2 = (const float*)d_in[21];
  const float* tWf1 = (const float*)d_in[22];
  const float* tbf1 = (const float*)d_in[23];
  const float* tWf2 = (const float*)d_in[24];
  const float* tbf2 = (const float*)d_in[25];
  const float* hW1 = (const float*)d_in[26];
  const float* hb1 = (const float*)d_in[27];
  const float* hW2 = (const float*)d_in[28];
  const float* hb2 = (const float*)d_in[29];

  float* ws   = (float*)d_ws;
  float* dinv = ws;                                  // N
  float* Tb   = dinv + N_;                           // N*128 (scaled messages)
  float* Hb   = Tb + (size_t)N_ * D_;                // N*128 (activations)
  float* seq  = Hb + (size_t)N_ * D_;                // 512*128
  float* qkv  = seq + (size_t)G_ * D_;               // 512*384
  float* att  = qkv + (size_t)G_ * 3 * D_;           // 512*128
  float* ffn  = att + (size_t)G_ * D_;               // 512*512
  float* tmp  = ffn + (size_t)G_ * 4 * D_;           // 512*128

  // ---- degrees: deg = 1 (self loop) + in-degree; dinv = rsqrt(deg) ----
  k_fill<<<cdiv(N_, 256), 256, 0, stream>>>(dinv, 1.0f, N_);
  k_deg_count<<<cdiv(E_, 256), 256, 0, stream>>>(edst, dinv);
  k_rsqrt_inplace<<<cdiv(N_, 256), 256, 0, stream>>>(dinv, N_);

  // ---- GCN layer 1: FIN->HID (scalar transform, fused scaling) ----
  k_gcn1_transform<<<cdiv((long long)N_ * HID_, 256), 256, 0, stream>>>(
      x, gW1, dinv, Tb, Hb);
  k_edge_scatter<<<cdiv((long long)E_ * HID_, 256), 256, 0, stream>>>(
      Tb, esrc, edst, dinv, Hb, 6);
  k_node_post<<<N_, HID_, 0, stream>>>(Hb, gb1, ln1g, ln1b, HID_, 1);

  // ---- GCN layer 2: HID->HID (WMMA, fused scaling + self-loop init) ----
  {
    int tilesMd4 = (N_ / 16) / 4, tilesN = HID_ / 16;
    k_wmma_gemm<0, 0, 0, 1, HID_><<<tilesMd4 * tilesN, 128, HID_ * 16 * 4, stream>>>(
        Hb, gW2, nullptr, dinv, Tb, Hb, tilesMd4, HID_, HID_, HID_);
  }
  k_edge_scatter<<<cdiv((long long)E_ * HID_, 256), 256, 0, stream>>>(
      Tb, esrc, edst, dinv, Hb, 6);
  k_node_post<<<N_, HID_, 0, stream>>>(Hb, gb2, ln2g, ln2b, HID_, 1);

  // ---- GCN layer 3: HID->D (WMMA, fused; bias added post-aggregation) ----
  {
    int tilesMd4 = (N_ / 16) / 4, tilesN = D_ / 16;
    k_wmma_gemm<0, 0, 0, 1, HID_><<<tilesMd4 * tilesN, 128, HID_ * 16 * 4, stream>>>(
        Hb, gW3, nullptr, dinv, Tb, Hb, tilesMd4, D_, HID_, D_);
  }
  k_edge_scatter<<<cdiv((long long)E_ * D_, 256), 256, 0, stream>>>(
      Tb, esrc, edst, dinv, Hb, 7);
  k_node_post<<<N_, D_, 0, stream>>>(Hb, gb3, nullptr, nullptr, D_, 0);

  // ---- readout mean + positional encoding ----
  k_readout_pe<<<G_, D_, 0, stream>>>(Hb, seq);

  // ---- transformer encoder layers ----
  int tMd4 = (G_ / 16) / 4;   // 8
  for (int l = 0; l < L_; ++l) {
    const float* Wqkv = tWqkv + (size_t)l * 3 * D_ * D_;
    const float* bqkv = tbqkv + (size_t)l * 3 * D_;
    const float* Wo   = tWo   + (size_t)l * D_ * D_;
    const float* bo   = tbo   + (size_t)l * D_;
    const float* Wf1  = tWf1  + (size_t)l * 4 * D_ * D_;
    const float* bf1  = tbf1  + (size_t)l * 4 * D_;
    const float* Wf2  = tWf2  + (size_t)l * D_ * 4 * D_;
    const float* bf2  = tbf2  + (size_t)l * D_;

    // qkv = seq @ Wqkv^T + bqkv   (512 x 384)
    k_wmma_gemm<1, 0, 1, 0, D_><<<tMd4 * (3 * D_ / 16), 128, D_ * 16 * 4, stream>>>(
        seq, Wqkv, bqkv, nullptr, qkv, nullptr, tMd4, 3 * D_, D_, 3 * D_);
    // multi-head attention -> att (512 x 128)
    k_attn<<<B_ * H_, T_, 0, stream>>>(qkv, att);
    // tmp = att @ Wo^T + bo
    k_wmma_gemm<1, 0, 1, 0, D_><<<tMd4 * (D_ / 16), 128, D_ * 16 * 4, stream>>>(
        att, Wo, bo, nullptr, tmp, nullptr, tMd4, D_, D_, D_);
    k_resid_ln<<<G_, D_, 0, stream>>>(seq, tmp, tg1 + l * D_, tb1 + l * D_);
    // ffn = relu(seq @ Wf1^T + bf1)   (512 x 512)
    k_wmma_gemm<1, 1, 1, 0, D_><<<tMd4 * (4 * D_ / 16), 128, D_ * 16 * 4, stream>>>(
        seq, Wf1, bf1, nullptr, ffn, nullptr, tMd4, 4 * D_, D_, 4 * D_);
    // tmp = ffn @ Wf2^T + bf2   (K = 512 -> 32KB LDS panel)
    k_wmma_gemm<1, 0, 1, 0, 4 * D_><<<tMd4 * (D_ / 16), 128, 4 * D_ * 16 * 4, stream>>>(
        ffn, Wf2, bf2, nullptr, tmp, nullptr, tMd4, D_, 4 * D_, D_);
    k_resid_ln<<<G_, D_, 0, stream>>>(seq, tmp, tg2 + l * D_, tb2 + l * D_);
  }

  // ---- pool + head ----
  k_head<<<B_, D_, 0, stream>>>(seq, hW1, hb1, hW2, hb2, (float*)d_out);
}